// NoisyBaseNode_25701084299362
// MI455X (gfx1250) — compile-verified
//
#include <hip/hip_runtime.h>

// NoisyLIF scan, T=5, B=32768, N=512.
// Memory-bound streaming kernel: 3 NT b128 loads + 1 NT b128 store per thread
// per timestep; membrane potential carried in registers across the unrolled
// time loop. WMMA is inapplicable (no contraction dimension).

#define T_STEPS  5
#define BATCH    32768
#define NUM_NODE 512
#define NV4      (NUM_NODE / 4)          // 128 vec4 per row

// SIGMA = 0.5 / sqrt(512)
#define SIGMA 0.022097086912079608f

// Native clang vector type: accepted by __builtin_nontemporal_load/store,
// layout-identical to 4 consecutive floats.
typedef __attribute__((ext_vector_type(4))) float v4f;

__global__ __launch_bounds__(256) void noisy_lif_scan_kernel(
    const v4f* __restrict__ x,    // [T, B, N/4]
    const v4f* __restrict__ nv,   // [B, T, N/4]
    const v4f* __restrict__ ns,   // [B, T, N/4]
    v4f*       __restrict__ y)    // [T, B, N/4]
{
    const unsigned g = blockIdx.x * blockDim.x + threadIdx.x;   // [0, B*NV4)
    const unsigned b = g >> 7;               // g / 128
    const unsigned c = g & (NV4 - 1);        // g % 128

    const long long xBase  = (long long)b * NV4 + c;               // + t * B*NV4
    const long long cnBase = (long long)b * (T_STEPS * NV4) + c;   // + t * NV4

    v4f v = (v4f)(0.0f);

#define LIF_STEP(comp)                                                        \
    do {                                                                      \
        float vi = 0.5f * (v.comp + __builtin_fmaf(SIGMA, nvt.comp, xt.comp));\
        const bool sp = (vi >= 0.5f);                                         \
        yt.comp = __builtin_fmaf(SIGMA, nst.comp, sp ? 1.0f : 0.0f);          \
        v.comp  = sp ? 0.0f : vi;                                             \
    } while (0)

#pragma unroll
    for (int t = 0; t < T_STEPS; ++t) {
        const long long xo  = xBase  + (long long)t * (BATCH * NV4);
        const long long cno = cnBase + t * NV4;

        const v4f xt  = __builtin_nontemporal_load(&x[xo]);
        const v4f nvt = __builtin_nontemporal_load(&nv[cno]);
        const v4f nst = __builtin_nontemporal_load(&ns[cno]);

        v4f yt;
        LIF_STEP(x);
        LIF_STEP(y);
        LIF_STEP(z);
        LIF_STEP(w);

        __builtin_nontemporal_store(yt, &y[xo]);
    }
#undef LIF_STEP
}

extern "C" void kernel_launch(void* const* d_in, const int* in_sizes, int n_in,
                              void* d_out, int out_size, void* d_ws, size_t ws_size,
                              hipStream_t stream) {
    (void)in_sizes; (void)n_in; (void)out_size; (void)d_ws; (void)ws_size;

    const v4f* x  = (const v4f*)d_in[0];  // x_seq [T, B, N]
    const v4f* nv = (const v4f*)d_in[1];  // cn_v  [B, T, N]
    const v4f* ns = (const v4f*)d_in[2];  // cn_s  [B, T, N]
    v4f*       y  = (v4f*)d_out;          // y     [T, B, N]

    const int totalThreads = BATCH * NV4;       // 4,194,304
    const int block = 256;                      // 8 wave32
    const int grid  = totalThreads / block;     // 16384

    noisy_lif_scan_kernel<<<grid, block, 0, stream>>>(x, nv, ns, y);
}